// CRFLayer_45174466019470
// MI455X (gfx1250) — compile-verified
//
#include <hip/hip_runtime.h>
#include <hip/hip_bf16.h>

typedef __attribute__((ext_vector_type(2))) float v2f;
typedef __attribute__((ext_vector_type(8))) float v8f;

#define CRF_B 128
#define CRF_S 2048
#define CRF_L 128
#define IGNORE_IDX (-100)
#define NEG_BIG (-3.402823466e38f)
#define ASTR 130  // padded LDS row strides (floats): avoid 16-way bank conflicts
#define PSTR 130
#define ESTR 130
#define LOG2E 1.4426950408889634f
#define LN2 0.6931471805599453f

// ---------------------------------------------------------------------------
// Kernel 1: gold-path score (one block of 128 threads per batch)
// ---------------------------------------------------------------------------
__global__ __launch_bounds__(128) void crf_score_kernel(
    const float* __restrict__ em, const long long* __restrict__ tags,
    const float* __restrict__ T, const float* __restrict__ startT,
    const float* __restrict__ endT, float* __restrict__ score_out) {
  const int b = blockIdx.x;
  const int tid = threadIdx.x;  // 0..127
  const long long* tb = tags + (long long)b * CRF_S;

  float partial = 0.0f;
  int local_last = -1;
  for (int s = tid; s < CRF_S; s += 128) {
    long long tg = tb[s];
    bool valid = (tg != IGNORE_IDX);
    if (valid && s > local_last) local_last = s;
    if (s >= 1 && valid) {
      long long pg = tb[s - 1];
      int prev = (pg == IGNORE_IDX) ? 0 : (int)pg;
      int cur = (int)tg;
      partial += T[prev * CRF_L + cur] +
                 em[((long long)b * CRF_S + s) * CRF_L + cur];
    }
  }

  __shared__ float red[128];
  __shared__ int redi[128];
  red[tid] = partial;
  redi[tid] = local_last;
  __syncthreads();
  for (int off = 64; off > 0; off >>= 1) {
    if (tid < off) {
      red[tid] += red[tid + off];
      redi[tid] = max(redi[tid], redi[tid + off]);
    }
    __syncthreads();
  }
  if (tid == 0) {
    float sc = red[0];
    long long t0 = tb[0];
    if (t0 != IGNORE_IDX) sc += startT[(int)t0];
    int last = redi[0];
    if (last >= 0) {
      long long lt = tb[last];
      sc += endT[(int)lt];
    }
    score_out[b] = sc;
  }
}

// ---------------------------------------------------------------------------
// Kernel 2: forward algorithm via exp-space WMMA matvec scan.
// 8 blocks x 256 threads (8 waves; one wave per 16-col N-tile of one 16-batch
// M-tile). Et = exp(T)^T LDS-resident. Per step: shfl row-max (no barriers)
// -> p = exp2 -> early em/mask/old-alpha loads -> barrier -> 32x
// v_wmma_f32_16x16x4_f32 -> branchless fma-blend masked update -> barrier.
// ---------------------------------------------------------------------------
__global__ __launch_bounds__(256) void crf_forward_kernel(
    const float* __restrict__ em, const unsigned char* __restrict__ mask,
    const float* __restrict__ T, const float* __restrict__ startT,
    const float* __restrict__ endT, float* __restrict__ logz_out) {
  __shared__ __align__(16) float Et[CRF_L * ESTR];   // exp(T)^T: Et[n][k]
  __shared__ __align__(16) float alpha[16 * ASTR];   // log-alphas
  __shared__ __align__(16) float pbuf[16 * PSTR];    // exp(alpha - rowmax)
  __shared__ __align__(16) float mrow[16];

  const int tid = threadIdx.x;   // 0..255
  const int lane = tid & 31;
  const int wave = tid >> 5;     // 0..7
  const int bb = blockIdx.x * 16;

  const int arow = lane & 15;    // M row of A-frag / N col of B- and D-frags
  const int ahalf = lane >> 4;   // 0: K pair {k0,k0+1}, 1: {k0+2,k0+3}
  const int nbase = wave * 16;
  const int col = nbase + arow;  // this lane's output column

  // Et[n][k] = exp(T[k][n]); alpha0 = startT + em[:,0,:]
  for (int i = tid; i < CRF_L * CRF_L; i += 256) {
    int k = i >> 7, n = i & (CRF_L - 1);
    Et[n * ESTR + k] = expf(T[i]);
  }
  for (int i = tid; i < 16 * CRF_L; i += 256) {
    int r = i >> 7, c = i & (CRF_L - 1);
    alpha[r * ASTR + c] = startT[c] + em[((long long)(bb + r) * CRF_S) * CRF_L + c];
  }
  __syncthreads();

  const int row = tid >> 4;   // 0..15: the alpha row this thread helps reduce
  const int sub = tid & 15;
  const float* prow = &pbuf[arow * PSTR];
  const float* brow = &Et[col * ESTR];
  // Per-lane bases: em element (bb+m, t, col) = embase + m*S*L + t*L
  const float* embase = em + ((long long)bb * CRF_S) * CRF_L + col;
  const unsigned char* mkbase = mask + (long long)bb * CRF_S;
  // Prefetch base: lane l prefetches row l%16, cols [nbase, nbase+16)
  const float* pfbase =
      em + (((long long)(bb + arow)) * CRF_S) * CRF_L + nbase;

  for (int t = 1; t < CRF_S; ++t) {
    // ---- one prefetch: 32 lanes cover all 16 rows of this wave's em tile ----
    __builtin_prefetch(pfbase + (long long)t * CRF_L, 0, 0);

    // ---- per-row max via wave-local shuffle (16 lanes own one row) ----
    float mx = NEG_BIG;
#pragma unroll
    for (int k = 0; k < 8; ++k)
      mx = fmaxf(mx, alpha[row * ASTR + sub + (k << 4)]);
    mx = fmaxf(mx, __shfl_xor(mx, 8));
    mx = fmaxf(mx, __shfl_xor(mx, 4));
    mx = fmaxf(mx, __shfl_xor(mx, 2));
    mx = fmaxf(mx, __shfl_xor(mx, 1));
    if (sub == 0) mrow[row] = mx;

    // ---- p = exp2((alpha - rowmax)*log2e): raw v_exp_f32, no denorm fixup ----
#pragma unroll
    for (int k = 0; k < 8; ++k) {
      int c = sub + (k << 4);
      pbuf[row * PSTR + c] =
          __builtin_amdgcn_exp2f((alpha[row * ASTR + c] - mx) * LOG2E);
    }

    // ---- early-issue this step's global loads + old alpha (hide under WMMA) --
    float ev[8], oldv[8], msel[8];
#pragma unroll
    for (int r = 0; r < 8; ++r) {
      int m = r + (ahalf << 3);
      ev[r] = embase[((long long)m * CRF_S + t) * CRF_L];
      msel[r] = (float)mkbase[(long long)m * CRF_S + t];  // 0.0 or 1.0
      oldv[r] = alpha[m * ASTR + col];
    }
    __syncthreads();

    // ---- q(16x16) = p(16x128) x E(128x128) via WMMA f32 16x16x4 ----
    v8f acc = {};
#pragma unroll 8
    for (int k0 = 0; k0 < CRF_L; k0 += 4) {
      int ak = k0 + (ahalf << 1);
      v2f a = *(const v2f*)(prow + ak);
      v2f bf = *(const v2f*)(brow + ak);
      acc = __builtin_amdgcn_wmma_f32_16x16x4_f32(
          false, a, false, bf, (short)0, acc, false, false);
    }

    // ---- rowmax values: two b128 LDS loads instead of 8 scalar ----
    float4 mr0 = *(const float4*)&mrow[(ahalf << 3)];
    float4 mr1 = *(const float4*)&mrow[(ahalf << 3) + 4];
    float mr[8] = {mr0.x, mr0.y, mr0.z, mr0.w, mr1.x, mr1.y, mr1.z, mr1.w};

    // ---- branchless masked update: log feeds an FMA (cannot if-convert) ----
#pragma unroll
    for (int r = 0; r < 8; ++r) {
      int m = r + (ahalf << 3);
      float newa = __builtin_amdgcn_logf(acc[r]) * LN2 + mr[r] + ev[r];
      alpha[m * ASTR + col] = fmaf(msel[r], newa - oldv[r], oldv[r]);
    }
    __syncthreads();  // alpha fully updated before next step's max
  }

  // ---- log_z = logsumexp(alpha + endT) per row: same shfl-reduce pattern ----
  {
    float mxf = NEG_BIG;
    float av[8];
#pragma unroll
    for (int k = 0; k < 8; ++k) {
      int c = sub + (k << 4);
      av[k] = alpha[row * ASTR + c] + endT[c];
      mxf = fmaxf(mxf, av[k]);
    }
    mxf = fmaxf(mxf, __shfl_xor(mxf, 8));
    mxf = fmaxf(mxf, __shfl_xor(mxf, 4));
    mxf = fmaxf(mxf, __shfl_xor(mxf, 2));
    mxf = fmaxf(mxf, __shfl_xor(mxf, 1));
    float s = 0.0f;
#pragma unroll
    for (int k = 0; k < 8; ++k) s += expf(av[k] - mxf);
    s += __shfl_xor(s, 8);
    s += __shfl_xor(s, 4);
    s += __shfl_xor(s, 2);
    s += __shfl_xor(s, 1);
    if (sub == 0) logz_out[bb + row] = mxf + logf(s);
  }
}

// ---------------------------------------------------------------------------
// Kernel 3: mean(log_z - score)
// ---------------------------------------------------------------------------
__global__ __launch_bounds__(128) void crf_final_kernel(
    const float* __restrict__ logz, const float* __restrict__ score,
    float* __restrict__ out) {
  __shared__ float red[128];
  int tid = threadIdx.x;
  red[tid] = logz[tid] - score[tid];
  __syncthreads();
  for (int off = 64; off > 0; off >>= 1) {
    if (tid < off) red[tid] += red[tid + off];
    __syncthreads();
  }
  if (tid == 0) out[0] = red[0] * (1.0f / CRF_B);
}

extern "C" void kernel_launch(void* const* d_in, const int* in_sizes, int n_in,
                              void* d_out, int out_size, void* d_ws,
                              size_t ws_size, hipStream_t stream) {
  (void)in_sizes; (void)n_in; (void)out_size; (void)ws_size;
  const float* em = (const float*)d_in[0];
  const unsigned char* mask = (const unsigned char*)d_in[1];
  const long long* tags = (const long long*)d_in[2];
  const float* T = (const float*)d_in[3];
  const float* startT = (const float*)d_in[4];
  const float* endT = (const float*)d_in[5];

  float* score = (float*)d_ws;            // 128 floats
  float* logz = (float*)d_ws + CRF_B;     // 128 floats
  float* out = (float*)d_out;

  crf_score_kernel<<<CRF_B, 128, 0, stream>>>(em, tags, T, startT, endT, score);
  crf_forward_kernel<<<CRF_B / 16, 256, 0, stream>>>(em, mask, T, startT, endT,
                                                     logz);
  crf_final_kernel<<<1, 128, 0, stream>>>(logz, score, out);
}